// _Interactions_22969485099927
// MI455X (gfx1250) — compile-verified
//
#include <hip/hip_runtime.h>
#include <hip/hip_bf16.h>
#include <stdint.h>

typedef __attribute__((ext_vector_type(2))) float v2f;
typedef __attribute__((ext_vector_type(8))) float v8f;

#define CH 64
#define NEG_SLOPE 0.2f

// ---------- order-preserving float <-> uint key (for atomic segment-max) ----------
__device__ __forceinline__ unsigned fkey(float f) {
    unsigned b = __float_as_uint(f);
    return (b & 0x80000000u) ? ~b : (b | 0x80000000u);
}
__device__ __forceinline__ float keyf(unsigned k) {
    unsigned b = (k & 0x80000000u) ? (k & 0x7FFFFFFFu) : ~k;
    return __uint_as_float(b);
}
#define KEY_NEG_INF 0x007FFFFFu  // fkey(-inf)

// ---------- fp32 WMMA GEMM: C[nrows x 64] = A[nrows x 64] * B[64 x 64] (+bias, opt softplus) ----------
// Block = 128 threads = 4 waves. Block computes a 16-row x 64-col tile; wave w owns cols [16w,16w+16).
// K-loop: 16 steps of V_WMMA_F32_16X16X4_F32.
// A frag (16x4 f32): lanes 0-15 -> M=lane, K={k0,k0+1}; lanes 16-31 -> M=lane-16, K={k0+2,k0+3}.
// B frag (4x16 f32): v0 = row K=k0 (lanes 0-15) / K=k0+2 (lanes 16-31); v1 = rows k0+1 / k0+3.
// C/D (16x16 f32): VGPR r: lanes 0-15 -> (M=r, N=lane); lanes 16-31 -> (M=r+8, N=lane-16).
template <bool SOFTPLUS>
__global__ void gemm64_wmma(const float* __restrict__ A,
                            const float* __restrict__ B,
                            const float* __restrict__ bias,
                            float* __restrict__ C,
                            int nrows) {
    const int lane = threadIdx.x & 31;
    const int wave = threadIdx.x >> 5;
    const int row0 = blockIdx.x * 16;
    if (row0 >= nrows) return;                 // uniform per block: EXEC stays all-1s for WMMA

    const int m    = lane & 15;                // row (A) / col (B,C) within tile
    const int half = lane >> 4;                // 0 or 1
    const int col0 = wave * 16;

    const float* __restrict__ arow = A + (size_t)(row0 + m) * CH + half * 2;
    const float* __restrict__ bcol = B + (size_t)(half * 2) * CH + col0 + m;

    v8f acc = {};
#pragma unroll
    for (int k0 = 0; k0 < CH; k0 += 4) {
        v2f a, b;
        a.x = arow[k0];
        a.y = arow[k0 + 1];
        b.x = bcol[(size_t)k0 * CH];
        b.y = bcol[(size_t)(k0 + 1) * CH];
        acc = __builtin_amdgcn_wmma_f32_16x16x4_f32(
            /*neg_a=*/false, a, /*neg_b=*/false, b,
            /*c_mod=*/(short)0, acc, /*reuse_a=*/false, /*reuse_b=*/false);
    }

    const int n  = col0 + m;
    const float bv = bias ? bias[n] : 0.0f;
#pragma unroll
    for (int r = 0; r < 8; ++r) {
        float v = acc[r] + bv;
        if (SOFTPLUS) v = (v > 20.0f) ? v : __logf(1.0f + __expf(v));
        C[(size_t)(row0 + half * 8 + r) * CH + n] = v;
    }
}

// ---------- per-node attention dots: a_src[n] = hw[n,:].att_src, a_dst[n] = hw[n,:].att_dst ----------
__global__ void att_dots(const float* __restrict__ hw,
                         const float* __restrict__ att_src,
                         const float* __restrict__ att_dst,
                         float* __restrict__ a_src,
                         float* __restrict__ a_dst,
                         int n) {
    const int node = (int)((blockIdx.x * (size_t)blockDim.x + threadIdx.x) >> 5);
    const int lane = threadIdx.x & 31;
    if (node >= n) return;                     // uniform per wave
    const float* row = hw + (size_t)node * CH;
    const float r0 = row[lane], r1 = row[lane + 32];
    float s = r0 * att_src[lane] + r1 * att_src[lane + 32];
    float d = r0 * att_dst[lane] + r1 * att_dst[lane + 32];
#pragma unroll
    for (int off = 16; off > 0; off >>= 1) {
        s += __shfl_xor(s, off, 32);
        d += __shfl_xor(d, off, 32);
    }
    if (lane == 0) { a_src[node] = s; a_dst[node] = d; }
}

// ---------- init accumulator (residual + bias) and softmax state ----------
__global__ void node_init(const float* __restrict__ h,
                          const float* __restrict__ bias,
                          float* __restrict__ accum,
                          unsigned* __restrict__ emax,
                          float* __restrict__ denom,
                          int n) {
    const size_t idx = blockIdx.x * (size_t)blockDim.x + threadIdx.x;
    if (idx >= (size_t)n * CH) return;
    const int node = (int)(idx >> 6);
    const int c    = (int)(idx & 63);
    accum[idx] = h[idx] + bias[c];
    if (c == 0) { emax[node] = KEY_NEG_INF; denom[node] = 0.0f; }
}

// ---------- edge pass 1: leaky-relu logits + atomic segment max over dst ----------
__global__ void edge_logits(const float* __restrict__ a_src,
                            const float* __restrict__ a_dst,
                            const long long* __restrict__ ei,
                            float* __restrict__ ebuf,
                            unsigned* __restrict__ emax,
                            long long E) {
    const long long e = blockIdx.x * (long long)blockDim.x + threadIdx.x;
    if (e >= E) return;
    const int s = (int)ei[e];
    const int d = (int)ei[E + e];
    float v = a_src[s] + a_dst[d];
    v = (v > 0.0f) ? v : NEG_SLOPE * v;
    ebuf[e] = v;
    atomicMax(&emax[d], fkey(v));
}

// ---------- edge pass 2: exp(e - max[dst]) + atomic segment sum ----------
__global__ void edge_exp(const float* __restrict__ ebuf,
                         const unsigned* __restrict__ emax,
                         const long long* __restrict__ ei,
                         float* __restrict__ eexp,
                         float* __restrict__ denom,
                         long long E) {
    const long long e = blockIdx.x * (long long)blockDim.x + threadIdx.x;
    if (e >= E) return;
    const int d = (int)ei[E + e];
    const float ex = __expf(ebuf[e] - keyf(emax[d]));
    eexp[e] = ex;
    atomicAdd(&denom[d], ex);
}

// ---------- edge pass 3: accum[dst] += (eexp/denom[dst]) * hw[src]  (wave per edge) ----------
__global__ void edge_scatter(const float* __restrict__ hw,
                             const float* __restrict__ eexp,
                             const float* __restrict__ denom,
                             const long long* __restrict__ ei,
                             float* __restrict__ accum,
                             long long E) {
    const long long e = (blockIdx.x * (long long)blockDim.x + threadIdx.x) >> 5;
    const int lane = threadIdx.x & 31;
    if (e >= E) return;                        // uniform per wave
    const int s = (int)ei[e];
    const int d = (int)ei[E + e];
    const float alpha = eexp[e] / denom[d];
    const float2 v = *(const float2*)(hw + (size_t)s * CH + 2 * lane);
    float* dp = accum + (size_t)d * CH + 2 * lane;
    atomicAdd(dp,     alpha * v.x);
    atomicAdd(dp + 1, alpha * v.y);
}

extern "C" void kernel_launch(void* const* d_in, const int* in_sizes, int n_in,
                              void* d_out, int out_size, void* d_ws, size_t ws_size,
                              hipStream_t stream) {
    (void)n_in; (void)out_size; (void)ws_size;
    const float*     x       = (const float*)d_in[0];
    // d_in[1] edge_weight, d_in[2] edge_attr: unused by the reference
    const float*     lin0_W  = (const float*)d_in[3];
    const float*     lin0_b  = (const float*)d_in[4];
    const float*     conv_W  = (const float*)d_in[5];   // [2,64,64]
    const float*     att_src = (const float*)d_in[6];   // [2,64]
    const float*     att_dst = (const float*)d_in[7];   // [2,64]
    const float*     conv_b  = (const float*)d_in[8];   // [2,64]
    const long long* ei      = (const long long*)d_in[9]; // [2,E] int64
    float* out = (float*)d_out;

    const int       N = in_sizes[0] / CH;
    const long long E = in_sizes[9] / 2;

    // workspace layout (floats)
    float* ws = (float*)d_ws;
    size_t off = 0;
    float*    h0    = ws + off; off += (size_t)N * CH;
    float*    hw    = ws + off; off += (size_t)N * CH;
    float*    acc1  = ws + off; off += (size_t)N * CH;
    float*    a_src = ws + off; off += (size_t)N;
    float*    a_dst = ws + off; off += (size_t)N;
    unsigned* emax  = (unsigned*)(ws + off); off += (size_t)N;
    float*    denom = ws + off; off += (size_t)N;
    float*    ebuf  = ws + off; off += (size_t)E;
    float*    eexp  = ws + off; off += (size_t)E;

    const int gemm_blocks = (N + 15) / 16;
    const int eblk  = (int)((E + 255) / 256);
    const int esblk = (int)((E * 32 + 255) / 256);

    // h0 = softplus(x @ lin0_W + lin0_b)
    gemm64_wmma<true><<<gemm_blocks, 128, 0, stream>>>(x, lin0_W, lin0_b, h0, N);

    for (int i = 0; i < 2; ++i) {
        const float* h     = (i == 0) ? h0   : acc1;
        float*       accum = (i == 0) ? acc1 : out;

        gemm64_wmma<false><<<gemm_blocks, 128, 0, stream>>>(h, conv_W + (size_t)i * CH * CH,
                                                            nullptr, hw, N);
        att_dots<<<(N + 7) / 8, 256, 0, stream>>>(hw, att_src + i * CH, att_dst + i * CH,
                                                  a_src, a_dst, N);
        node_init<<<(int)(((size_t)N * CH + 255) / 256), 256, 0, stream>>>(h, conv_b + i * CH,
                                                                           accum, emax, denom, N);
        edge_logits<<<eblk, 256, 0, stream>>>(a_src, a_dst, ei, ebuf, emax, E);
        edge_exp<<<eblk, 256, 0, stream>>>(ebuf, emax, ei, eexp, denom, E);
        edge_scatter<<<esblk, 256, 0, stream>>>(hw, eexp, denom, ei, accum, E);
    }
}